// GraphSAGE_91276644974851
// MI455X (gfx1250) — compile-verified
//
#include <hip/hip_runtime.h>
#include <hip/hip_bf16.h>

// GraphSAGE 2-layer forward for MI455X (gfx1250, wave32).
// Strategy: project-then-aggregate (mean is linear) to halve scatter bytes;
// fp32 WMMA (v_wmma_f32_16x16x4_f32) for both GEMMs with LDS-staged weights;
// global f32 atomics for the edge scatter; wave32 shuffle log_softmax.

typedef __attribute__((ext_vector_type(2))) float v2f;
typedef __attribute__((ext_vector_type(8))) float v8f;

// ---------------------------------------------------------------- utilities

__global__ void sage_zero_kernel(float* __restrict__ p, long long n) {
    long long i = (long long)blockIdx.x * blockDim.x + threadIdx.x;
    if (i < n) p[i] = 0.0f;
}

__global__ void sage_degree_kernel(const int* __restrict__ dst,
                                   float* __restrict__ deg, int nEdges) {
    int e = blockIdx.x * blockDim.x + threadIdx.x;
    if (e < nEdges) atomicAdd(&deg[dst[e]], 1.0f);
}

// ------------------------------------------------------------- WMMA GEMM
// Y[M, OUT_A+OUT_B] = X[M,K] @ concat(Wa[OUT_A,K], Wb[OUT_B,K])^T
// One wave computes a 16-row x OUT-col stripe; fp32 matrix core:
// D(16x16) = A(16x4) x B(4x16) + C, K advanced 4 per WMMA.
//
// Weights are staged once per block into LDS (row stride K+4 floats so the
// 16-lane b64 fragment reads are bank-conflict free), so B fragments come
// from ds_load_b64 instead of global round trips. Fragment loads for a
// K-step are batched into distinct registers before the WMMAs so they
// overlap instead of serializing on a shared register pair.
//
// A fragment (32-bit 16x4 per ISA): lane&15 -> row M, lane>>4 selects K-pair;
//   VGPR0 = K=(k0 + 2*(lane>>4)), VGPR1 = +1. B fragment mirrors with N=lane&15.
// C/D (32-bit 16x16): lane&15 -> N, VGPR v -> M = v + 8*(lane>>4).
template <int K, int OUT_A, int OUT_B>
__global__ void sage_gemm_wmma_kernel(const float* __restrict__ X,
                                      const float* __restrict__ Wa,
                                      const float* __restrict__ Wb,
                                      float* __restrict__ Y, int nRows) {
    constexpr int OUT  = OUT_A + OUT_B;
    constexpr int NT   = OUT / 16;          // 16-wide column tiles
    constexpr int LSTR = K + 4;             // padded LDS row stride (floats)

    __shared__ float wlds[OUT * LSTR];

    // --- cooperative weight stage (all waves participate, then barrier) ---
    for (int idx = threadIdx.x; idx < OUT * K; idx += blockDim.x) {
        const int o = idx / K;
        const int k = idx - o * K;
        const float w = (o < OUT_A) ? Wa[(size_t)o * K + k]
                                    : Wb[(size_t)(o - OUT_A) * K + k];
        wlds[o * LSTR + k] = w;
    }
    __syncthreads();

    const int lane    = threadIdx.x & 31;
    const int waveId  = blockIdx.x * (blockDim.x >> 5) + (threadIdx.x >> 5);
    const int rowBase = waveId * 16;
    if (rowBase >= nRows) return;           // whole wave exits; EXEC all-1s

    const int half = lane >> 4;             // 0 or 1
    const int l15  = lane & 15;
    const int row  = rowBase + l15;         // A-fragment row for this lane

    v8f acc[NT];
#pragma unroll
    for (int t = 0; t < NT; ++t) { v8f z = {}; acc[t] = z; }

    const float* __restrict__ Xrow = X + (size_t)row * K;

    for (int k0 = 0; k0 < K; k0 += 4) {
        const int kk = k0 + half * 2;       // even -> 8B aligned b64 loads

        // batch all fragment loads for this K-step
        const v2f a = *(const v2f*)(Xrow + kk);
        v2f b[NT];
#pragma unroll
        for (int t = 0; t < NT; ++t) {
            const int col = t * 16 + l15;
            b[t] = *(const v2f*)(&wlds[col * LSTR + kk]);
        }
        // then issue the WMMAs
#pragma unroll
        for (int t = 0; t < NT; ++t) {
            acc[t] = __builtin_amdgcn_wmma_f32_16x16x4_f32(
                false, a, false, b[t], (short)0, acc[t], false, false);
        }
    }

#pragma unroll
    for (int t = 0; t < NT; ++t) {
#pragma unroll
        for (int v = 0; v < 8; ++v) {
            const int m = v + half * 8;
            Y[(size_t)(rowBase + m) * OUT + t * 16 + l15] = acc[t][v];
        }
    }
}

// ------------------------------------------------------------- edge scatter
// Layer 1: 64 message channels live in y1[:, 0:64] (row stride 128).
// One edge per wave; lane handles a float2 -> 64 global_atomic_add_f32/edge.
__global__ void sage_scatter64_kernel(const float* __restrict__ y1,
                                      const int* __restrict__ src,
                                      const int* __restrict__ dst,
                                      float* __restrict__ agg, int nEdges) {
    const int wave = blockIdx.x * (blockDim.x >> 5) + (threadIdx.x >> 5);
    const int lane = threadIdx.x & 31;
    if (wave >= nEdges) return;
    const int s = src[wave];
    const int d = dst[wave];
    const v2f m = *(const v2f*)(y1 + (size_t)s * 128 + lane * 2);
    float* p = agg + (size_t)d * 64 + lane * 2;
    atomicAdd(p,     m.x);
    atomicAdd(p + 1, m.y);
}

// Layer 2: 32 message channels in y2[:, 0:32] (row stride 64); one float/lane.
__global__ void sage_scatter32_kernel(const float* __restrict__ y2,
                                      const int* __restrict__ src,
                                      const int* __restrict__ dst,
                                      float* __restrict__ agg, int nEdges) {
    const int wave = blockIdx.x * (blockDim.x >> 5) + (threadIdx.x >> 5);
    const int lane = threadIdx.x & 31;
    if (wave >= nEdges) return;
    const int s = src[wave];
    const int d = dst[wave];
    const float m = y2[(size_t)s * 64 + lane];
    atomicAdd(agg + (size_t)d * 32 + lane, m);
}

// ------------------------------------------------------------- epilogues

// h = relu(agg1/max(deg,1) + b1 + self1), self1 = y1[:, 64:128]
__global__ void sage_finish1_kernel(const float* __restrict__ agg1,
                                    const float* __restrict__ deg,
                                    const float* __restrict__ y1,
                                    const float* __restrict__ b1,
                                    float* __restrict__ h, long long n64) {
    long long idx = (long long)blockIdx.x * blockDim.x + threadIdx.x;
    if (idx >= n64) return;
    const long long i = idx >> 6;
    const int       c = (int)(idx & 63);
    const float inv = 1.0f / fmaxf(deg[i], 1.0f);
    const float v = agg1[idx] * inv + b1[c] + y1[i * 128 + 64 + c];
    h[idx] = fmaxf(v, 0.0f);
}

// out = log_softmax(agg2/max(deg,1) + b2 + self2); one wave32 per node,
// lane == class index (32 classes). self2 = y2[:, 32:64].
__global__ void sage_finish2_kernel(const float* __restrict__ agg2,
                                    const float* __restrict__ deg,
                                    const float* __restrict__ y2,
                                    const float* __restrict__ b2,
                                    float* __restrict__ out, int nNodes) {
    const int node = blockIdx.x * (blockDim.x >> 5) + (threadIdx.x >> 5);
    const int lane = threadIdx.x & 31;
    if (node >= nNodes) return;
    const float inv = 1.0f / fmaxf(deg[node], 1.0f);
    float v = agg2[(size_t)node * 32 + lane] * inv + b2[lane]
            + y2[(size_t)node * 64 + 32 + lane];
    float mx = v;
#pragma unroll
    for (int off = 16; off > 0; off >>= 1) mx = fmaxf(mx, __shfl_xor(mx, off, 32));
    float e = expf(v - mx);
#pragma unroll
    for (int off = 16; off > 0; off >>= 1) e += __shfl_xor(e, off, 32);
    out[(size_t)node * 32 + lane] = (v - mx) - logf(e);
}

// ---------------------------------------------------------------- launcher

extern "C" void kernel_launch(void* const* d_in, const int* in_sizes, int n_in,
                              void* d_out, int out_size, void* d_ws, size_t ws_size,
                              hipStream_t stream) {
    const float* x   = (const float*)d_in[0];
    const int*   ei  = (const int*)  d_in[1];   // [2, E] (src row, dst row)
    const float* W1l = (const float*)d_in[2];   // [64,128]
    const float* b1  = (const float*)d_in[3];   // [64]
    const float* W1r = (const float*)d_in[4];   // [64,128]
    const float* W2l = (const float*)d_in[5];   // [32,64]
    const float* b2  = (const float*)d_in[6];   // [32]
    const float* W2r = (const float*)d_in[7];   // [32,64]
    float* out = (float*)d_out;

    const int N = in_sizes[0] / 128;            // 100000 (divisible by 16)
    const int E = in_sizes[1] / 2;              // 1600000
    const int* src = ei;
    const int* dst = ei + E;

    // Workspace layout (floats): [deg N][agg1 N*64][h N*64][y N*128]
    // agg2 aliases agg1[0:N*32]; y2 aliases y[0:N*64] (safe: same stream order).
    float* ws   = (float*)d_ws;
    float* deg  = ws;
    float* agg1 = ws + N;
    float* hbuf = agg1 + (size_t)N * 64;
    float* ybuf = hbuf + (size_t)N * 64;
    float* agg2 = agg1;

    const int B = 256;                          // 8 wave32 per block
    const int wavesPerBlk = B / 32;

    // 1) zero deg + agg1 (contiguous region)
    {
        long long n = (long long)N + (long long)N * 64;
        sage_zero_kernel<<<(unsigned)((n + B - 1) / B), B, 0, stream>>>(ws, n);
    }
    // 2) degrees
    sage_degree_kernel<<<(E + B - 1) / B, B, 0, stream>>>(dst, deg, E);
    // 3) y1 = x @ [W1_l ; W1_r]^T   ([N,128], cols 0:64 = msg, 64:128 = self)
    {
        int waves = (N + 15) / 16;
        sage_gemm_wmma_kernel<128, 64, 64>
            <<<(waves + wavesPerBlk - 1) / wavesPerBlk, B, 0, stream>>>(
                x, W1l, W1r, ybuf, N);
    }
    // 4) agg1[dst] += msg1[src]
    sage_scatter64_kernel<<<(E + wavesPerBlk - 1) / wavesPerBlk, B, 0, stream>>>(
        ybuf, src, dst, agg1, E);
    // 5) h = relu(agg1/deg + b1 + self1)
    {
        long long n = (long long)N * 64;
        sage_finish1_kernel<<<(unsigned)((n + B - 1) / B), B, 0, stream>>>(
            agg1, deg, ybuf, b1, hbuf, n);
    }
    // 6) zero agg2 (aliases agg1[0:N*32])
    {
        long long n = (long long)N * 32;
        sage_zero_kernel<<<(unsigned)((n + B - 1) / B), B, 0, stream>>>(agg2, n);
    }
    // 7) y2 = h @ [W2_l ; W2_r]^T   ([N,64], cols 0:32 = msg, 32:64 = self)
    {
        int waves = (N + 15) / 16;
        sage_gemm_wmma_kernel<64, 32, 32>
            <<<(waves + wavesPerBlk - 1) / wavesPerBlk, B, 0, stream>>>(
                hbuf, W2l, W2r, ybuf, N);
    }
    // 8) agg2[dst] += msg2[src]
    sage_scatter32_kernel<<<(E + wavesPerBlk - 1) / wavesPerBlk, B, 0, stream>>>(
        ybuf, src, dst, agg2, E);
    // 9) log_softmax epilogue -> d_out
    sage_finish2_kernel<<<(N + wavesPerBlk - 1) / wavesPerBlk, B, 0, stream>>>(
        agg2, deg, ybuf, b2, out, N);
}